// BasicAttention_8160437862501
// MI455X (gfx1250) — compile-verified
//
#include <hip/hip_runtime.h>
#include <hip/hip_bf16.h>
#include <stdint.h>

// ---------------- problem constants (from reference) ----------------
#define BATCH      2048
#define NTOK       49          // N = 7*7
#define DIMK       384         // DIM
#define NHEAD      8
#define KEYD       32          // KEY_DIM
#define DV         128         // ATTN_RATIO * KEY_DIM
#define HCOLS      192         // per-head qkv cols: 32 q + 32 k + 128 v
#define DH         1024        // NHEAD * DV
#define QROWS      52          // token rows padded to mult of 4 (K-dim of P@V)
#define SCALE      0.17677669529663687f   // 32^-0.5

typedef float v2f __attribute__((ext_vector_type(2)));
typedef float v8f __attribute__((ext_vector_type(8)));
typedef unsigned int u32x4 __attribute__((ext_vector_type(4)));
typedef int i32x4 __attribute__((ext_vector_type(4)));
typedef int i32x8 __attribute__((ext_vector_type(8)));

__device__ __forceinline__ v8f wmma_f32(v2f a, v2f b, v8f c) {
  // D = A(16x4, f32) * B(4x16, f32) + C(16x16, f32)
  return __builtin_amdgcn_wmma_f32_16x16x4_f32(
      /*neg_a=*/false, a, /*neg_b=*/false, b,
      /*c_mod=*/(short)0, c, /*reuse_a=*/false, /*reuse_b=*/false);
}

// ---------------------------------------------------------------------
// Tensor Data Mover: DMA a 2-D f32 tile (tile_d1 rows x tile_d0 cols,
// row stride stride0 elements) from global memory into LDS, packed
// row-major at lds_addr.  D# layout per CDNA5 ISA ch.8 (groups 0/1;
// groups 2/3 zero for <=2D tensors).  Issue from ONE wave, then
// s_wait_tensorcnt 0 + workgroup barrier.
// ---------------------------------------------------------------------
__device__ __forceinline__ void tdm_load_2d_f32(unsigned int lds_addr,
                                                const float* gptr,
                                                unsigned int tensor_d0,
                                                unsigned int tensor_d1,
                                                unsigned int tile_d0,
                                                unsigned int tile_d1,
                                                unsigned int stride0) {
  const unsigned long long ga = (unsigned long long)(uintptr_t)gptr;
  u32x4 g0;
  g0[0] = 1u;                                   // count=1, user mode, no gather
  g0[1] = lds_addr;                             // LDS byte address
  g0[2] = (unsigned int)ga;                     // global_addr[31:0]
  g0[3] = (unsigned int)((ga >> 32) & 0x01FFFFFFu) | 0x80000000u; // [56:32] | type=2
  i32x8 g1;
  g1[0] = (int)(2u << 16);                      // data_size = 4 bytes
  g1[1] = (int)((tensor_d0 & 0xFFFFu) << 16);   // tensor_dim0[15:0] @ bits63:48
  g1[2] = (int)(((tensor_d0 >> 16) & 0xFFFFu) | ((tensor_d1 & 0xFFFFu) << 16));
  g1[3] = (int)(((tensor_d1 >> 16) & 0xFFFFu) | ((tile_d0 & 0xFFFFu) << 16));
  g1[4] = (int)(tile_d1 & 0xFFFFu);             // tile_dim1 (tile_dim2 = 0)
  g1[5] = (int)stride0;                         // tensor_dim0_stride[31:0]
  g1[6] = 0;
  g1[7] = 0;
  i32x4 gz = {0, 0, 0, 0};
#if __has_include(<hip/amd_detail/amd_gfx1250_TDM.h>)
  i32x8 gz8 = {0, 0, 0, 0, 0, 0, 0, 0};         // therock clang-23: 6-arg form
  __builtin_amdgcn_tensor_load_to_lds(g0, g1, gz, gz, gz8, 0);
#else
  __builtin_amdgcn_tensor_load_to_lds(g0, g1, gz, gz, 0);  // ROCm 7.2: 5-arg
#endif
}

// =====================================================================
// Kernel 1: fused  qkv-projection (head slice) + biased attention
// grid.x = BATCH*NHEAD, block = 128 threads (4 waves)
// =====================================================================
__global__ __launch_bounds__(128)
void levit_qkv_attn(const float* __restrict__ x,
                    const float* __restrict__ qkv_w,
                    const float* __restrict__ qkv_b,
                    const float* __restrict__ attn_bias,
                    const int*   __restrict__ bias_idxs,
                    float* __restrict__ attn_out,
                    int n_off) {
  const int bh   = blockIdx.x;
  const int b    = bh >> 3;
  const int h    = bh & 7;
  const int tid  = threadIdx.x;
  const int wave = tid >> 5;          // 0..3 -> M tile
  const int lane = tid & 31;
  const int half = lane >> 4;         // 0 | 1
  const int l16  = lane & 15;

  __shared__ __align__(16) float lds_qkv[QROWS * HCOLS]; // 52x192 f32 (39.9 KB)
  // x-chunk staging (stage A) aliases the score/prob buffer (stages B/C)
  __shared__ __align__(16) union {
    float xc[NTOK * 64];              // 12.5 KB : TDM-staged x[b][:, kc*64 .. +64)
    float p[64 * 64];                 // 16   KB : scores / probabilities
  } u;

  // zero the (padded) qkv buffer so stale rows never feed a K-reduction
  for (int i = tid; i < QROWS * HCOLS; i += 128) lds_qkv[i] = 0.0f;

  const float* xb    = x + (size_t)b * NTOK * DIMK;
  const int    hbase = h * HCOLS;

  // ---------------- Stage A: qkv_h = x[b] @ W_h^T + b_h -------------
  {
    const int mt   = wave;
    const int arow = mt * 16 + l16;                       // A row (M)
    // per-lane weight base; the 12 N-tiles sit at constant byte strides
    const float* wp = qkv_w + (size_t)(hbase + l16) * DIMK + (half ? 2 : 0);
    v8f acc[12] = {};
    for (int kc = 0; kc < 6; ++kc) {                      // 6 K-chunks of 64
      __syncthreads();                                    // chunk buffer free?
      if (tid < 32) {                                     // wave 0 drives the TDM
        tdm_load_2d_f32((unsigned int)(uintptr_t)u.xc, xb + kc * 64,
                        /*tensor_d0=*/DIMK, /*tensor_d1=*/NTOK,
                        /*tile_d0=*/64, /*tile_d1=*/NTOK, /*stride0=*/DIMK);
        __builtin_amdgcn_s_wait_tensorcnt(0);
      }
      __syncthreads();                                    // chunk visible to all
      for (int kk = 0; kk < 64; kk += 4) {
        const int k0 = kk + (half ? 2 : 0);
        v2f a;
        if (arow < NTOK) {
          a.x = u.xc[arow * 64 + k0];
          a.y = u.xc[arow * 64 + k0 + 1];
        } else {
          a.x = 0.0f; a.y = 0.0f;
        }
        const int kg = kc * 64 + k0;
#pragma unroll
        for (int nt = 0; nt < 12; ++nt) {
          v2f bb;
          bb.x = wp[nt * 16 * DIMK + kg];                 // imm-offset loads
          bb.y = wp[nt * 16 * DIMK + kg + 1];
          acc[nt] = wmma_f32(a, bb, acc[nt]);
        }
      }
    }
#pragma unroll
    for (int nt = 0; nt < 12; ++nt) {
      const float qb = qkv_b[hbase + nt * 16 + l16];
#pragma unroll
      for (int r = 0; r < 8; ++r) {
        const int tok = mt * 16 + half * 8 + r;           // C/D: M = r + 8*half
        if (tok < NTOK)
          lds_qkv[tok * HCOLS + nt * 16 + l16] = acc[nt][r] + qb;
      }
    }
  }
  __syncthreads();

  // ---------------- Stage B: S = q k^T * scale + bias ---------------
  {
    const int mt   = wave;
    const int qrow = mt * 16 + l16;
    for (int nt = 0; nt < 4; ++nt) {
      const int kcol = nt * 16 + l16;
      v8f acc = {};
#pragma unroll
      for (int kk = 0; kk < KEYD; kk += 4) {
        const int k0 = kk + (half ? 2 : 0);
        v2f a, bb;
        if (qrow < NTOK) {
          a.x = lds_qkv[qrow * HCOLS + k0];
          a.y = lds_qkv[qrow * HCOLS + k0 + 1];
        } else { a.x = 0.0f; a.y = 0.0f; }
        if (kcol < NTOK) {
          bb.x = lds_qkv[kcol * HCOLS + KEYD + k0];
          bb.y = lds_qkv[kcol * HCOLS + KEYD + k0 + 1];
        } else { bb.x = 0.0f; bb.y = 0.0f; }
        acc = wmma_f32(a, bb, acc);
      }
#pragma unroll
      for (int r = 0; r < 8; ++r) {
        const int qr = mt * 16 + half * 8 + r;
        const int kc = nt * 16 + l16;
        float s = 0.0f;
        if (qr < NTOK && kc < NTOK)
          s = acc[r] * SCALE + attn_bias[h * n_off + bias_idxs[qr * NTOK + kc]];
        u.p[qr * 64 + kc] = s;
      }
    }
  }
  __syncthreads();

  // ---------------- softmax over the 49 keys of each query row ------
  if (tid < NTOK) {
    const int r = tid;
    float m = -1e30f;
    for (int c = 0; c < NTOK; ++c) m = fmaxf(m, u.p[r * 64 + c]);
    float s = 0.0f;
    for (int c = 0; c < NTOK; ++c) s += __expf(u.p[r * 64 + c] - m);
    const float inv = 1.0f / s;
    for (int c = 0; c < NTOK; ++c)
      u.p[r * 64 + c] = __expf(u.p[r * 64 + c] - m) * inv;
    // cols 49..63 already hold exact zeros from Stage B
  }
  __syncthreads();

  // ---------------- Stage C: out_h = P @ V --------------------------
  {
    const int mt   = wave;
    const int qrow = mt * 16 + l16;
    for (int nt = 0; nt < 8; ++nt) {
      v8f acc = {};
#pragma unroll
      for (int kk = 0; kk < QROWS; kk += 4) {   // K padded to 52; P cols >=49 == 0
        const int k0 = kk + (half ? 2 : 0);
        v2f a, bb;
        a.x = u.p[qrow * 64 + k0];
        a.y = u.p[qrow * 64 + k0 + 1];
        const int vcol = 2 * KEYD + nt * 16 + l16;  // v slice starts at col 64
        bb.x = lds_qkv[k0 * HCOLS + vcol];          // padded v rows are zero
        bb.y = lds_qkv[(k0 + 1) * HCOLS + vcol];
        acc = wmma_f32(a, bb, acc);
      }
#pragma unroll
      for (int r = 0; r < 8; ++r) {
        const int tok = mt * 16 + half * 8 + r;
        if (tok < NTOK)
          attn_out[((size_t)b * NTOK + tok) * DH + h * DV + nt * 16 + l16] = acc[r];
      }
    }
  }
}

// =====================================================================
// Kernel 2: out = attn_out (M x 1024) @ proj_w^T (1024 x 384) + proj_b
// grid.x = M/16 = 6272, block = 128 threads (4 waves)
// =====================================================================
__global__ __launch_bounds__(128)
void levit_proj(const float* __restrict__ ao,
                const float* __restrict__ proj_w,
                const float* __restrict__ proj_b,
                float* __restrict__ out) {
  const int tid  = threadIdx.x;
  const int wave = tid >> 5;
  const int lane = tid & 31;
  const int half = lane >> 4;
  const int l16  = lane & 15;
  const size_t r0 = (size_t)blockIdx.x * 16;

  __shared__ __align__(16) float lds_a[16 * 256];  // one K-chunk of A (16 KB)

  // per-lane weight base; the 6 N-tiles sit at constant byte strides
  const float* pw = proj_w + (size_t)(wave * 16 + l16) * DH + (half ? 2 : 0);

  v8f acc[6] = {};                                  // wave covers 6 of 24 N-tiles
  for (int kc = 0; kc < 4; ++kc) {
    __syncthreads();                                // chunk buffer free?
    if (tid < 32) {                                 // wave 0 drives the TDM
      tdm_load_2d_f32((unsigned int)(uintptr_t)lds_a, ao + r0 * DH + kc * 256,
                      /*tensor_d0=*/DH, /*tensor_d1=*/16,
                      /*tile_d0=*/256, /*tile_d1=*/16, /*stride0=*/DH);
      __builtin_amdgcn_s_wait_tensorcnt(0);
    }
    // warm L1/L2 for this chunk of the weight stream (global_prefetch_b8)
    __builtin_prefetch(pw + kc * 256, 0, 1);
    __syncthreads();                                // chunk visible to all

    for (int kk = 0; kk < 256; kk += 4) {
      const int k0 = kk + (half ? 2 : 0);
      v2f a;
      a.x = lds_a[l16 * 256 + k0];
      a.y = lds_a[l16 * 256 + k0 + 1];
      const int kg = kc * 256 + k0;
#pragma unroll
      for (int j = 0; j < 6; ++j) {
        v2f bb;
        bb.x = pw[(size_t)j * 64 * DH + kg];        // imm-offset loads
        bb.y = pw[(size_t)j * 64 * DH + kg + 1];
        acc[j] = wmma_f32(a, bb, acc[j]);
      }
    }
  }

#pragma unroll
  for (int j = 0; j < 6; ++j) {
    const int col = (wave + j * 4) * 16 + l16;
    const float pb = proj_b[col];
#pragma unroll
    for (int r = 0; r < 8; ++r) {
      const size_t row = r0 + half * 8 + r;
      out[row * DIMK + col] = acc[j][r] + pb;
    }
  }
}

// =====================================================================
extern "C" void kernel_launch(void* const* d_in, const int* in_sizes, int n_in,
                              void* d_out, int out_size, void* d_ws, size_t ws_size,
                              hipStream_t stream) {
  (void)n_in; (void)out_size; (void)ws_size;
  const float* x         = (const float*)d_in[0];
  const float* qkv_w     = (const float*)d_in[1];
  const float* qkv_b     = (const float*)d_in[2];
  const float* proj_w    = (const float*)d_in[3];
  const float* proj_b    = (const float*)d_in[4];
  const float* attn_bias = (const float*)d_in[5];
  const int*   bias_idxs = (const int*)d_in[6];
  float*       out       = (float*)d_out;
  float*       attn_out  = (float*)d_ws;           // (B*N) x 1024 f32 = 411 MB

  const int n_off = in_sizes[5] / NHEAD;           // 49 unique offsets

  levit_qkv_attn<<<BATCH * NHEAD, 128, 0, stream>>>(
      x, qkv_w, qkv_b, attn_bias, bias_idxs, attn_out, n_off);

  levit_proj<<<(BATCH * NTOK) / 16, 128, 0, stream>>>(
      attn_out, proj_w, proj_b, out);
}